// PPCNodeLayer_81930796139195
// MI455X (gfx1250) — compile-verified
//
#include <hip/hip_runtime.h>
#include <hip/hip_bf16.h>
#include <math.h>

// ---------------------------------------------------------------------------
// Problem constants (from reference)
// ---------------------------------------------------------------------------
#define BB 4
#define TT 1024
#define DD 1024
#define EE 8
#define NN (BB * TT)          // 4096 tokens
#define D2 (2 * DD)           // 2048 features
#define CC (2 * NN / EE)      // 1024 capacity per expert
#define LOCAL_ITERS 8

typedef float v2f __attribute__((ext_vector_type(2)));
typedef float v8f __attribute__((ext_vector_type(8)));

// ---------------------------------------------------------------------------
// 1) Init: x_states = x_stream copy; x_target = [x[:,0], rotate(x[:,:-1])]
//    layout: x_flat[n][2d+k] == x_stream[b,t,d,k]
// ---------------------------------------------------------------------------
__global__ __launch_bounds__(256)
void k_init(const float* __restrict__ x_stream,
            const float* __restrict__ cos_p,
            const float* __restrict__ sin_p,
            float* __restrict__ xs, float* __restrict__ xt) {
    int id = blockIdx.x * 256 + threadIdx.x;          // (b*T + t)*D + d
    if (id >= NN * DD) return;
    int d  = id & (DD - 1);
    int bt = id >> 10;                                 // b*T + t
    int t  = bt & (TT - 1);
    size_t cur = (size_t)id * 2;

    float2 v = *(const float2*)(x_stream + cur);
    *(float2*)(xs + cur) = v;

    float2 tg;
    if (t == 0) {
        tg = v;
    } else {
        size_t prev = ((size_t)(bt - 1) * DD + d) * 2;
        float2 p = *(const float2*)(x_stream + prev);
        float c = cos_p[d], s = sin_p[d];
        tg.x = p.x * c - p.y * s;
        tg.y = p.x * s + p.y * c;
    }
    *(float2*)(xt + cur) = tg;
}

// ---------------------------------------------------------------------------
// 2) inv map fill (-1)
// ---------------------------------------------------------------------------
__global__ __launch_bounds__(256)
void k_fill_inv(int* __restrict__ inv) {
    int id = blockIdx.x * 256 + threadIdx.x;
    if (id < EE * NN) inv[id] = -1;
}

// ---------------------------------------------------------------------------
// 3) Gating: probs[n][e] = softmax(x[n] @ Wg)[e].  One wave per token.
// ---------------------------------------------------------------------------
__global__ __launch_bounds__(256)
void k_gating(const float* __restrict__ xs, const float* __restrict__ Wg,
              float* __restrict__ probs) {
    int wid  = (blockIdx.x * 256 + threadIdx.x) >> 5;  // token
    int lane = threadIdx.x & 31;
    if (wid >= NN) return;
    const float* xrow = xs + (size_t)wid * D2;

    float acc[EE];
#pragma unroll
    for (int e = 0; e < EE; ++e) acc[e] = 0.f;

    for (int k = lane; k < D2; k += 32) {
        float xv = xrow[k];
        const float* wrow = Wg + (size_t)k * EE;
#pragma unroll
        for (int e = 0; e < EE; ++e) acc[e] += xv * wrow[e];
    }
#pragma unroll
    for (int e = 0; e < EE; ++e) {
#pragma unroll
        for (int off = 16; off > 0; off >>= 1)
            acc[e] += __shfl_xor(acc[e], off, 32);
    }
    if (lane == 0) {
        float m = acc[0];
#pragma unroll
        for (int e = 1; e < EE; ++e) m = fmaxf(m, acc[e]);
        float ex[EE], s = 0.f;
#pragma unroll
        for (int e = 0; e < EE; ++e) { ex[e] = expf(acc[e] - m); s += ex[e]; }
        float is = 1.f / s;
#pragma unroll
        for (int e = 0; e < EE; ++e) probs[(size_t)wid * EE + e] = ex[e] * is;
    }
}

// ---------------------------------------------------------------------------
// 4) Expert-choice top-C per expert: bitonic sort of (prob, tie-break) keys.
//    One block per expert, 512 threads, 4096 u64 keys in LDS (32 KB).
// ---------------------------------------------------------------------------
__global__ __launch_bounds__(512)
void k_topk(const float* __restrict__ probs, float* __restrict__ scores,
            int* __restrict__ indices, int* __restrict__ inv) {
    __shared__ unsigned long long keys[NN];
    const int e   = blockIdx.x;
    const int tid = threadIdx.x;

    for (int n = tid; n < NN; n += 512) {
        unsigned pb = __float_as_uint(probs[(size_t)n * EE + e]); // p >= 0 -> monotonic
        keys[n] = ((unsigned long long)pb << 32) | (unsigned)(0xFFFFFFFFu - n);
    }
    __syncthreads();

    for (int size = 2; size <= NN; size <<= 1) {
        for (int stride = size >> 1; stride > 0; stride >>= 1) {
            for (int t = tid; t < NN / 2; t += 512) {
                int i = 2 * t - (t & (stride - 1));
                int j = i + stride;
                unsigned long long a = keys[i], b = keys[j];
                bool descBlock = ((i & size) == 0);
                if (descBlock ? (a < b) : (a > b)) { keys[i] = b; keys[j] = a; }
            }
            __syncthreads();
        }
    }

    for (int c = tid; c < CC; c += 512) {
        unsigned long long k = keys[c];
        unsigned n = 0xFFFFFFFFu - (unsigned)(k & 0xFFFFFFFFull);
        scores[e * CC + c]  = __uint_as_float((unsigned)(k >> 32));
        indices[e * CC + c] = (int)n;
        inv[e * NN + (int)n] = c;
    }
}

// ---------------------------------------------------------------------------
// 5) Expert GEMM with fp32 WMMA:
//    Y[e][c][:] = gather(xs, indices[e]) @ We[e] + be[e]
//    grid = (Ntiles=8, Mtiles=16, E=8), block = 256 (8 waves).
//    Block tile 64x256, K-slab 32.  Waves arranged 2(M) x 4(N), wave tile
//    32x64 -> 8x v8f accumulators -> 8 WMMAs per K-step of 4.
//    B is staged in LDS in fragment-native layout:
//      Bsm[g*1024 + n*4 + kq] = B[k0 + 4g + kq][n0 + n]
//    so each lane's (b.x,b.y) pair is one aligned ds_load_b64.
// ---------------------------------------------------------------------------
__global__ __launch_bounds__(256)
void k_expert_gemm(const float* __restrict__ xs, const float* __restrict__ We,
                   const float* __restrict__ be, const int* __restrict__ indices,
                   float* __restrict__ Y) {
    constexpr int LDA = 36;                 // 32 + 4 pad
    __shared__ __align__(16) float Asm[64 * LDA];      //  9 KB
    __shared__ __align__(16) float Bsm[8 * 256 * 4];   // 32 KB
    __shared__ int idxs[64];

    const int e     = blockIdx.z;
    const int mtile = blockIdx.y;   // 0..15 (rows of C, 64 each)
    const int ntile = blockIdx.x;   // 0..7  (cols of D2, 256 each)
    const int n0    = ntile * 256;
    const int tid   = threadIdx.x;
    const int wave  = tid >> 5;
    const int lane  = tid & 31;
    const int wy    = wave & 1;     // 2 waves along M (2*32 = 64)
    const int wx    = wave >> 1;    // 4 waves along N (4*64 = 256)
    const int r     = lane & 15;
    const int kh    = lane >> 4;

    if (tid < 64) idxs[tid] = indices[e * CC + mtile * 64 + tid];
    __syncthreads();

    v8f acc[2][4] = {};

    for (int k0 = 0; k0 < D2; k0 += 32) {
        // ---- Stage A: 64 gathered rows x 32 cols (512 float4 / 256 thr) ----
#pragma unroll
        for (int it = 0; it < 2; ++it) {
            int li = tid + it * 256;
            int ar = li >> 3;            // 0..63
            int cg = li & 7;             // 0..7 float4 group
            const float4 v = *(const float4*)(xs + (size_t)idxs[ar] * D2 + k0 + cg * 4);
            *(float4*)&Asm[ar * LDA + cg * 4] = v;
        }
        // ---- Stage B fragment-native: gather 4 k-rows per (g,n) quad ------
        // 2048 (g,n) pairs / 256 threads = 8 each; lanes -> consecutive n
        const float* WbBase = We + ((size_t)e * D2 + k0) * D2 + n0;
#pragma unroll
        for (int it = 0; it < 8; ++it) {
            int p = tid + it * 256;      // 0..2047
            int g = p >> 8;              // 0..7 k-group
            int n = p & 255;             // 0..255 col
            const float* src = WbBase + (size_t)(g * 4) * D2 + n;
            float4 v;
            v.x = src[0];
            v.y = src[D2];
            v.z = src[2 * D2];
            v.w = src[3 * D2];
            *(float4*)&Bsm[(g << 10) + (n << 2)] = v;
        }
        // ---- Prefetch next K-slab into cache while this one computes ------
        int nk = k0 + 32;
        if (nk < D2) {
            __builtin_prefetch(We + ((size_t)e * D2 + nk + (tid >> 3)) * D2 +
                                   n0 + (tid & 7) * 32, 0, 3);
            if (tid < 64)
                __builtin_prefetch(xs + (size_t)idxs[tid] * D2 + nk, 0, 3);
        }
        __syncthreads();

        // ---- Compute: 8x v_wmma_f32_16x16x4_f32 per K-step of 4 -----------
#pragma unroll
        for (int kk = 0; kk < 32; kk += 4) {
            const int g = kk >> 2;
            v2f af[2], bf[4];
#pragma unroll
            for (int mb = 0; mb < 2; ++mb) {
                // A 16x4 frag: lanes 0-15 -> K=kk+0/1 ; lanes 16-31 -> K=kk+2/3
                int row = wy * 32 + mb * 16 + r;
                af[mb].x = Asm[row * LDA + kk + kh * 2 + 0];
                af[mb].y = Asm[row * LDA + kk + kh * 2 + 1];
            }
#pragma unroll
            for (int nb = 0; nb < 4; ++nb) {
                // B 4x16 frag: one aligned b64 per lane
                int col = wx * 64 + nb * 16 + r;
                bf[nb] = *(const v2f*)&Bsm[(g << 10) + (col << 2) + kh * 2];
            }
#pragma unroll
            for (int mb = 0; mb < 2; ++mb)
#pragma unroll
                for (int nb = 0; nb < 4; ++nb)
                    acc[mb][nb] = __builtin_amdgcn_wmma_f32_16x16x4_f32(
                        false, af[mb], false, bf[nb], (short)0, acc[mb][nb],
                        false, false);
        }
        __syncthreads();
    }

    // Epilogue: C/D layout -> lanes0-15: M = v, N = lane ; lanes16-31: M = 8+v
    const float* berow = be + (size_t)e * D2;
#pragma unroll
    for (int mb = 0; mb < 2; ++mb) {
#pragma unroll
        for (int nb = 0; nb < 4; ++nb) {
            int ncol = n0 + wx * 64 + nb * 16 + r;
#pragma unroll
            for (int v = 0; v < 8; ++v) {
                int mrow = mtile * 64 + wy * 32 + mb * 16 + kh * 8 + v;
                Y[((size_t)e * CC + mrow) * D2 + ncol] = acc[mb][nb][v] + berow[ncol];
            }
        }
    }
}

// ---------------------------------------------------------------------------
// 6) Update: xs += lr * clip(xt - pred, ±10); pred gathered per token (determ.)
// ---------------------------------------------------------------------------
__global__ __launch_bounds__(256)
void k_update(const float* __restrict__ xt, float* __restrict__ xs,
              const float* __restrict__ Y, const int* __restrict__ inv,
              const float* __restrict__ scores, float lr) {
    __shared__ int   inv_s[EE];
    __shared__ float sc_s[EE];
    const int n   = blockIdx.x;
    const int tid = threadIdx.x;
    if (tid < EE) {
        int c = inv[tid * NN + n];
        inv_s[tid] = c;
        sc_s[tid]  = (c >= 0) ? scores[tid * CC + c] : 0.f;
    }
    __syncthreads();

#pragma unroll
    for (int jj = 0; jj < 8; ++jj) {
        int j = jj * 256 + tid;
        float pv = 0.f;
#pragma unroll
        for (int e = 0; e < EE; ++e) {
            int c = inv_s[e];
            if (c >= 0) pv += sc_s[e] * Y[((size_t)e * CC + c) * D2 + j];
        }
        size_t o = (size_t)n * D2 + j;
        float resid = xt[o] - pv;
        resid = fminf(10.f, fmaxf(-10.f, resid));
        xs[o] = xs[o] + lr * resid;
    }
}

// ---------------------------------------------------------------------------
// 7) Final: out = xs + 0.5*(xt - pred); per-token partial sum of |residual|_2
// ---------------------------------------------------------------------------
__global__ __launch_bounds__(256)
void k_final(const float* __restrict__ xt, const float* __restrict__ xs,
             const float* __restrict__ Y, const int* __restrict__ inv,
             const float* __restrict__ scores, float* __restrict__ out,
             float* __restrict__ partial) {
    __shared__ int   inv_s[EE];
    __shared__ float sc_s[EE];
    __shared__ float red[256];
    const int n   = blockIdx.x;
    const int tid = threadIdx.x;
    if (tid < EE) {
        int c = inv[tid * NN + n];
        inv_s[tid] = c;
        sc_s[tid]  = (c >= 0) ? scores[tid * CC + c] : 0.f;
    }
    __syncthreads();

    float lsum = 0.f;
#pragma unroll
    for (int dd = 0; dd < 4; ++dd) {
        int d = dd * 256 + tid;
        int j = 2 * d;
        float pr = 0.f, pi = 0.f;
#pragma unroll
        for (int e = 0; e < EE; ++e) {
            int c = inv_s[e];
            if (c >= 0) {
                float2 y = *(const float2*)(Y + ((size_t)e * CC + c) * D2 + j);
                pr += sc_s[e] * y.x;
                pi += sc_s[e] * y.y;
            }
        }
        size_t o = (size_t)n * D2 + j;
        float dr = xt[o] - pr;
        float di = xt[o + 1] - pi;
        out[o]     = xs[o]     + 0.5f * dr;
        out[o + 1] = xs[o + 1] + 0.5f * di;
        lsum += sqrtf(dr * dr + di * di);
    }
    red[tid] = lsum;
    __syncthreads();
#pragma unroll
    for (int s = 128; s > 0; s >>= 1) {
        if (tid < s) red[tid] += red[tid + s];
        __syncthreads();
    }
    if (tid == 0) partial[n] = red[0];
}

// ---------------------------------------------------------------------------
// 8) Deterministic reduce of 4096 partials + trailing scalar outputs
// ---------------------------------------------------------------------------
__global__ __launch_bounds__(256)
void k_reduce(const float* __restrict__ partial, float* __restrict__ out) {
    __shared__ float red[256];
    const int tid = threadIdx.x;
    float s = 0.f;
#pragma unroll
    for (int k = 0; k < NN / 256; ++k) s += partial[k * 256 + tid];
    red[tid] = s;
    __syncthreads();
#pragma unroll
    for (int st = 128; st > 0; st >>= 1) {
        if (tid < st) red[tid] += red[tid + st];
        __syncthreads();
    }
    if (tid == 0) {
        out[(size_t)NN * D2]     = (float)LOCAL_ITERS;
        out[(size_t)NN * D2 + 1] = red[0] / (float)(NN * DD);
    }
}

// ---------------------------------------------------------------------------
// Host launch
// ---------------------------------------------------------------------------
extern "C" void kernel_launch(void* const* d_in, const int* in_sizes, int n_in,
                              void* d_out, int out_size, void* d_ws, size_t ws_size,
                              hipStream_t stream) {
    const float* x_stream = (const float*)d_in[0];
    const float* cos_p    = (const float*)d_in[1];
    const float* sin_p    = (const float*)d_in[2];
    const float* Wg       = (const float*)d_in[3];
    const float* We       = (const float*)d_in[4];
    const float* be       = (const float*)d_in[5];
    float* out = (float*)d_out;

    // Workspace carve-up
    float* xs      = (float*)d_ws;                    // [N][2D]   32 MB
    float* xt      = xs + (size_t)NN * D2;            // [N][2D]   32 MB
    float* probs   = xt + (size_t)NN * D2;            // [N][E]
    float* scores  = probs + (size_t)NN * EE;         // [E][C]
    int*   indices = (int*)(scores + (size_t)EE * CC);// [E][C]
    int*   inv     = indices + (size_t)EE * CC;       // [E][N]
    float* Y       = (float*)(inv + (size_t)EE * NN); // [E][C][2D] 64 MB
    float* partial = Y + (size_t)EE * CC * D2;        // [N]

    k_init<<<(NN * DD) / 256, 256, 0, stream>>>(x_stream, cos_p, sin_p, xs, xt);

    float lr = 0.5f;
    for (int iter = 0; iter <= LOCAL_ITERS; ++iter) {
        k_fill_inv<<<(EE * NN) / 256, 256, 0, stream>>>(inv);
        k_gating<<<(NN * 32) / 256, 256, 0, stream>>>(xs, Wg, probs);
        k_topk<<<EE, 512, 0, stream>>>(probs, scores, indices, inv);
        k_expert_gemm<<<dim3(D2 / 256, CC / 64, EE), 256, 0, stream>>>(
            xs, We, be, indices, Y);
        if (iter < LOCAL_ITERS) {
            k_update<<<NN, 256, 0, stream>>>(xt, xs, Y, inv, scores, lr);
            lr *= 0.85f;
        } else {
            k_final<<<NN, 256, 0, stream>>>(xt, xs, Y, inv, scores, out, partial);
            k_reduce<<<1, 256, 0, stream>>>(partial, out);
        }
    }
}